// MolecularGNN_7636451852858
// MI455X (gfx1250) — compile-verified
//
#include <hip/hip_runtime.h>
#include <cstdint>
#include <cstddef>

// ---------------------------------------------------------------------------
// GCN forward for MI455X (gfx1250, wave32).
//  - fp32 WMMA 16x16x4 for the dense GEMMs (full precision, tensor pipe)
//  - edge scatter with native f32 global atomics (agg buffers fit in 192MB L2)
// ---------------------------------------------------------------------------

typedef __attribute__((ext_vector_type(2))) float v2f;
typedef __attribute__((ext_vector_type(8))) float v8f;

#define H      128
#define FIN    10
#define BN_EPS 1e-5f
#define APAD   132   // A-tile row pitch: banks (4m+k)%64 distinct for m=0..15
#define BPAD   144   // B-tile row pitch: rows k and k+2 land 32 banks apart
#define WPAD   80    // fc1 tile pitch (64 cols): rows k,k+2 32 banks apart

#if defined(__HIP_DEVICE_COMPILE__)
#if !__has_builtin(__builtin_amdgcn_wmma_f32_16x16x4_f32)
#error "f32 WMMA builtin missing on this toolchain (device pass)"
#endif
#endif

static __device__ __forceinline__ v8f wmma_f32(v2f a, v2f b, v8f c) {
  // 8 args: (neg_a, A, neg_b, B, c_mod, C, reuse_a, reuse_b)
  return __builtin_amdgcn_wmma_f32_16x16x4_f32(false, a, false, b, (short)0, c,
                                               false, false);
}

// -------------------------------- utility ---------------------------------

__global__ void k_fill(float* __restrict__ p, float v, int n) {
  int i = blockIdx.x * blockDim.x + threadIdx.x;
  if (i < n) p[i] = v;
}

// deg[dst] += 1 per edge (self-loop added in k_dinv)
__global__ void k_degree(const int* __restrict__ dst, float* __restrict__ deg, int e) {
  int i = blockIdx.x * blockDim.x + threadIdx.x;
  if (i < e) unsafeAtomicAdd(&deg[dst[i]], 1.0f);
}

__global__ void k_dinv(float* __restrict__ deg, int n) {
  int i = blockIdx.x * blockDim.x + threadIdx.x;
  if (i < n) deg[i] = rsqrtf(deg[i] + 1.0f);  // +1 self-loop; deg>=1 always
}

// ------------------------- layer 1 linear (K=10) ---------------------------

__global__ void __launch_bounds__(H) k_linear1(const float* __restrict__ x,
                                               const float* __restrict__ W1,
                                               float* __restrict__ h, int n) {
  __shared__ float xs[FIN];
  int node = blockIdx.x;
  int c = threadIdx.x;
  if (c < FIN) xs[c] = x[(size_t)node * FIN + c];
  __syncthreads();
  float acc = 0.f;
#pragma unroll
  for (int k = 0; k < FIN; ++k) acc = fmaf(xs[k], W1[k * H + c], acc);
  h[(size_t)node * H + c] = acc;
}

// ----------------------------- propagation ---------------------------------

// agg[i][c] = h[i][c]*dinv[i]^2 + bias[c]   (self-loop term + conv bias)
// float4-vectorized: 32 vec4 per node row.
__global__ void k_agg_init(const float4* __restrict__ h4, const float* __restrict__ dinv,
                           const float4* __restrict__ b4, float4* __restrict__ agg4,
                           int n) {
  int idx = blockIdx.x * blockDim.x + threadIdx.x;
  if (idx < n * (H / 4)) {
    int i = idx >> 5;
    int c4 = idx & 31;
    float di = dinv[i];
    float s = di * di;
    float4 hv = h4[idx], bv = b4[c4], o;
    o.x = fmaf(hv.x, s, bv.x);
    o.y = fmaf(hv.y, s, bv.y);
    o.z = fmaf(hv.z, s, bv.z);
    o.w = fmaf(hv.w, s, bv.w);
    agg4[idx] = o;
  }
}

// one wave per edge: 128 channels, float4 loads, 4 f32 atomics per lane.
__global__ void __launch_bounds__(256) k_scatter(const float* __restrict__ h,
                                                 const int* __restrict__ src,
                                                 const int* __restrict__ dst,
                                                 const float* __restrict__ dinv,
                                                 float* __restrict__ agg, int e) {
  int wid = (blockIdx.x * blockDim.x + threadIdx.x) >> 5;
  int lane = threadIdx.x & 31;
  if (wid >= e) return;
  if (lane == 0) {  // stream index arrays ahead (global_prefetch_b8)
    __builtin_prefetch(src + wid + 4096, 0, 1);
    __builtin_prefetch(dst + wid + 4096, 0, 1);
  }
  int s = src[wid], d = dst[wid];
  float coef = dinv[s] * dinv[d];
  const float4* hv = reinterpret_cast<const float4*>(h + (size_t)s * H) + lane;
  float4 v = *hv;
  float* ap = agg + (size_t)d * H + lane * 4;
  unsafeAtomicAdd(ap + 0, v.x * coef);
  unsafeAtomicAdd(ap + 1, v.y * coef);
  unsafeAtomicAdd(ap + 2, v.z * coef);
  unsafeAtomicAdd(ap + 3, v.w * coef);
}

// ------------------------------ batch norm ---------------------------------

__global__ void __launch_bounds__(256) k_bn_stats(const float* __restrict__ h,
                                                  float* __restrict__ stats, int n) {
  int c = threadIdx.x & (H - 1);
  int half = threadIdx.x >> 7;
  int i0 = blockIdx.x * 256 + half;
  int iend = blockIdx.x * 256 + 256;
  if (iend > n) iend = n;
  float s = 0.f, sq = 0.f;
  for (int i = i0; i < iend; i += 2) {
    float v = h[(size_t)i * H + c];
    s += v;
    sq += v * v;
  }
  unsafeAtomicAdd(&stats[c], s);
  unsafeAtomicAdd(&stats[H + c], sq);
}

__global__ void k_bn_finalize(const float* __restrict__ stats,
                              const float* __restrict__ gamma,
                              const float* __restrict__ beta,
                              float* __restrict__ ss, int n) {
  int c = threadIdx.x;  // 128 threads
  float inv_n = 1.0f / (float)n;
  float mean = stats[c] * inv_n;
  float var = fmaxf(stats[H + c] * inv_n - mean * mean, 0.f);
  float scale = gamma[c] * rsqrtf(var + BN_EPS);
  ss[c] = scale;
  ss[H + c] = beta[c] - mean * scale;
}

// float4-vectorized BN apply + ReLU. ss4[0..31]=scale vec4, ss4[32..63]=shift.
__global__ void k_bn_apply_relu(const float4* __restrict__ h4, const float4* __restrict__ ss4,
                                float4* __restrict__ o4, int n) {
  int idx = blockIdx.x * blockDim.x + threadIdx.x;
  if (idx < n * (H / 4)) {
    int c4 = idx & 31;
    float4 hv = h4[idx];
    float4 sc = ss4[c4];
    float4 sh = ss4[32 + c4];
    float4 o;
    o.x = fmaxf(fmaf(hv.x, sc.x, sh.x), 0.f);
    o.y = fmaxf(fmaf(hv.y, sc.y, sh.y), 0.f);
    o.z = fmaxf(fmaf(hv.z, sc.z, sh.z), 0.f);
    o.w = fmaxf(fmaf(hv.w, sc.w, sh.w), 0.f);
    o4[idx] = o;
  }
}

// --------------------- N x 128 @ 128 x 128 fp32 WMMA GEMM ------------------
// 256 threads = 8 waves; block computes 32 rows x 128 cols. Wave w owns cols
// [16w,16w+16) for BOTH 16-row tiles (two accumulators) so every B fragment
// feeds two WMMAs. K=128 staged through LDS in 4 phases of 32.
__global__ void __launch_bounds__(256) k_gemm_h_h(const float* __restrict__ A,
                                                  const float* __restrict__ B,
                                                  float* __restrict__ C, int n) {
  __shared__ float As[32 * APAD];
  __shared__ float Bs[32 * BPAD];
  int tid = threadIdx.x;
  int row0 = blockIdx.x * 32;

  for (int t = tid; t < 32 * H; t += 256) {
    int r = t >> 7, k = t & (H - 1);
    int gr = row0 + r;
    if (gr >= n) gr = n - 1;
    As[r * APAD + k] = A[(size_t)gr * H + k];
  }

  int wv = tid >> 5, lane = tid & 31;
  int laneM = lane & 15, hi = lane >> 4;
  int col0 = wv * 16;
  v8f acc0 = {}, acc1 = {};

  for (int ph = 0; ph < 4; ++ph) {
    __syncthreads();
    for (int t = tid; t < 32 * H; t += 256) {
      int r = t >> 7, c = t & (H - 1);
      Bs[r * BPAD + c] = B[(size_t)(ph * 32 + r) * H + c];
    }
    __syncthreads();
#pragma unroll
    for (int kq = 0; kq < 8; ++kq) {
      int kk = kq * 4 + 2 * hi;   // local K row for this half-wave
      int kg = ph * 32 + kk;      // global K
      v2f a0, a1, b;
      b.x = Bs[kk * BPAD + col0 + laneM];          // B[k][n], B[k+1][n]
      b.y = Bs[(kk + 1) * BPAD + col0 + laneM];
      a0.x = As[laneM * APAD + kg];                // rows 0..15
      a0.y = As[laneM * APAD + kg + 1];
      a1.x = As[(16 + laneM) * APAD + kg];         // rows 16..31
      a1.y = As[(16 + laneM) * APAD + kg + 1];
      acc0 = wmma_f32(a0, b, acc0);
      acc1 = wmma_f32(a1, b, acc1);
    }
  }

  int rbase = hi * 8;  // lane<16 -> M=r ; lane>=16 -> M=8+r (ISA C/D layout)
  if (row0 + 32 <= n) {  // fast path: unguarded stores (N % 32 == 0 case)
#pragma unroll
    for (int r = 0; r < 8; ++r) {
      C[(size_t)(row0 + rbase + r) * H + col0 + laneM] = acc0[r];
      C[(size_t)(row0 + 16 + rbase + r) * H + col0 + laneM] = acc1[r];
    }
  } else {
#pragma unroll
    for (int r = 0; r < 8; ++r) {
      int g0 = row0 + rbase + r, g1 = g0 + 16;
      if (g0 < n) C[(size_t)g0 * H + col0 + laneM] = acc0[r];
      if (g1 < n) C[(size_t)g1 * H + col0 + laneM] = acc1[r];
    }
  }
}

// ---------------- fused head: FC1(WMMA)+ReLU -> FC2 -> log_softmax ---------
// 128 threads = 4 waves; 16 nodes per block; 4 waves x 16 = 64 FC1 cols.
__global__ void __launch_bounds__(128) k_head(const float* __restrict__ A,
                                              const float* __restrict__ fw1,
                                              const float* __restrict__ fb1,
                                              const float* __restrict__ fw2,
                                              const float* __restrict__ fb2,
                                              float* __restrict__ out, int n) {
  __shared__ float As[16 * APAD];
  __shared__ float Ws[32 * WPAD];
  __shared__ float h3s[16 * 66];
  __shared__ float lg[32];
  int tid = threadIdx.x;
  int row0 = blockIdx.x * 16;

  for (int t = tid; t < 16 * H; t += 128) {
    int r = t >> 7, k = t & (H - 1);
    int gr = row0 + r;
    if (gr >= n) gr = n - 1;
    As[r * APAD + k] = A[(size_t)gr * H + k];
  }

  int wv = tid >> 5, lane = tid & 31;
  int laneM = lane & 15, hi = lane >> 4;
  int col0 = wv * 16;
  v8f acc = {};

  for (int ph = 0; ph < 4; ++ph) {
    __syncthreads();
    for (int t = tid; t < 32 * 64; t += 128) {
      int r = t >> 6, c = t & 63;
      Ws[r * WPAD + c] = fw1[(size_t)(ph * 32 + r) * 64 + c];
    }
    __syncthreads();
#pragma unroll
    for (int kq = 0; kq < 8; ++kq) {
      int kk = kq * 4 + 2 * hi;
      int kg = ph * 32 + kk;
      v2f a, b;
      a.x = As[laneM * APAD + kg];
      a.y = As[laneM * APAD + kg + 1];
      b.x = Ws[kk * WPAD + col0 + laneM];
      b.y = Ws[(kk + 1) * WPAD + col0 + laneM];
      acc = wmma_f32(a, b, acc);
    }
  }

  float bias = fb1[col0 + laneM];
  int rbase = hi * 8;
#pragma unroll
  for (int r = 0; r < 8; ++r)
    h3s[(rbase + r) * 66 + col0 + laneM] = fmaxf(acc[r] + bias, 0.f);
  __syncthreads();

  if (tid < 32) {  // 16 nodes x 2 classes
    int nd = tid >> 1, cls = tid & 1;
    float l = fb2[cls];
#pragma unroll
    for (int j = 0; j < 64; ++j) l = fmaf(h3s[nd * 66 + j], fw2[j * 2 + cls], l);
    lg[tid] = l;
  }
  __syncthreads();
  if (tid < 32) {
    int nd = tid >> 1, cls = tid & 1;
    int gr = row0 + nd;
    if (gr < n) {
      float l0 = lg[nd * 2], l1 = lg[nd * 2 + 1];
      float m = fmaxf(l0, l1);
      float lse = m + logf(expf(l0 - m) + expf(l1 - m));
      out[(size_t)gr * 2 + cls] = (cls ? l1 : l0) - lse;
    }
  }
}

// ------------------------------ launcher -----------------------------------

extern "C" void kernel_launch(void* const* d_in, const int* in_sizes, int n_in,
                              void* d_out, int out_size, void* d_ws, size_t ws_size,
                              hipStream_t stream) {
  const float* x   = (const float*)d_in[0];
  const int*   ei  = (const int*)d_in[1];
  const float* W1  = (const float*)d_in[2];
  const float* b1  = (const float*)d_in[3];
  const float* g1  = (const float*)d_in[4];
  const float* be1 = (const float*)d_in[5];
  const float* W2  = (const float*)d_in[6];
  const float* b2  = (const float*)d_in[7];
  const float* g2  = (const float*)d_in[8];
  const float* be2 = (const float*)d_in[9];
  const float* fw1 = (const float*)d_in[10];
  const float* fb1 = (const float*)d_in[11];
  const float* fw2 = (const float*)d_in[12];
  const float* fb2 = (const float*)d_in[13];
  float* out = (float*)d_out;

  int n = in_sizes[0] / FIN;   // 100000
  int e = in_sizes[1] / 2;     // 1600000
  const int* src = ei;         // edge_index[0]
  const int* dst = ei + e;     // edge_index[1]

  // workspace layout (fp32): dinv[n] | stats[256] | ss[256] | buf0[n*H] | buf1[n*H]
  float* ws    = (float*)d_ws;
  size_t nAl   = ((size_t)n + 255) & ~(size_t)255;
  float* dinv  = ws;
  float* stats = ws + nAl;
  float* ss    = stats + 2 * H;
  float* buf0  = ss + 2 * H;
  float* buf1  = buf0 + (size_t)n * H;

  int nv4 = n * (H / 4);                 // float4 elements in an N x H buffer
  int gNV4 = (nv4 + 255) / 256;
  int gN = (n + 255) / 256;
  int gE = (e + 255) / 256;
  int gEdgeWaves = (e + 7) / 8;          // 8 edges (waves) per 256-thread block
  int gemmBlocks = (n + 31) / 32;
  int headBlocks = (n + 15) / 16;

  // ---- normalization coefficients
  k_fill<<<gN, 256, 0, stream>>>(dinv, 0.f, n);
  k_degree<<<gE, 256, 0, stream>>>(dst, dinv, e);
  k_dinv<<<gN, 256, 0, stream>>>(dinv, n);

  // ---- GCN layer 1
  k_linear1<<<n, H, 0, stream>>>(x, W1, buf0, n);
  k_agg_init<<<gNV4, 256, 0, stream>>>((const float4*)buf0, dinv, (const float4*)b1,
                                       (float4*)buf1, n);
  k_scatter<<<gEdgeWaves, 256, 0, stream>>>(buf0, src, dst, dinv, buf1, e);
  k_fill<<<1, 256, 0, stream>>>(stats, 0.f, 2 * H);
  k_bn_stats<<<gN, 256, 0, stream>>>(buf1, stats, n);
  k_bn_finalize<<<1, H, 0, stream>>>(stats, g1, be1, ss, n);
  k_bn_apply_relu<<<gNV4, 256, 0, stream>>>((const float4*)buf1, (const float4*)ss,
                                            (float4*)buf0, n);

  // ---- GCN layer 2
  k_gemm_h_h<<<gemmBlocks, 256, 0, stream>>>(buf0, W2, buf1, n);
  k_agg_init<<<gNV4, 256, 0, stream>>>((const float4*)buf1, dinv, (const float4*)b2,
                                       (float4*)buf0, n);
  k_scatter<<<gEdgeWaves, 256, 0, stream>>>(buf1, src, dst, dinv, buf0, e);
  k_fill<<<1, 256, 0, stream>>>(stats, 0.f, 2 * H);
  k_bn_stats<<<gN, 256, 0, stream>>>(buf0, stats, n);
  k_bn_finalize<<<1, H, 0, stream>>>(stats, g2, be2, ss, n);
  k_bn_apply_relu<<<gNV4, 256, 0, stream>>>((const float4*)buf0, (const float4*)ss,
                                            (float4*)buf1, n);

  // ---- head
  k_head<<<headBlocks, 128, 0, stream>>>(buf1, fw1, fb1, fw2, fb2, out, n);

  (void)n_in; (void)out_size; (void)ws_size;
}